// TwoTierPreFilter_32255204393107
// MI455X (gfx1250) — compile-verified
//
#include <hip/hip_runtime.h>
#include <hip/hip_bf16.h>

typedef float v2f __attribute__((ext_vector_type(2)));
typedef float v4f __attribute__((ext_vector_type(4)));
typedef float v8f __attribute__((ext_vector_type(8)));

#define B_SZ   64
#define P_SZ   8192
#define CIN    22
#define CPAD   24
#define HC     128
#define HR     384
#define TOPN   600
#define OFF_V  1000000.0f
#define NEGINF (-__builtin_inff())

__device__ __forceinline__ void ttpf_wait_async0() {
#if __has_builtin(__builtin_amdgcn_s_wait_asynccnt)
    __builtin_amdgcn_s_wait_asynccnt(0);
#else
    asm volatile("s_wait_asynccnt 0" ::: "memory");
#endif
}

// ---------------- coarse MLP: 22 -> 128 -> 128 -> 1 over all points ----------------
#define NT        64             // points per block
#define NTILES    (NT / 16)      // 4 N-tiles per block
#define XS_STRIDE 28             // CPAD + 4 : lanes hit banks 4n -> conflict-free b64
#define HS_STRIDE (HC + 4)       // 132 : 16B-aligned rows, conflict-free b64/b128

__global__ __launch_bounds__(256) void ttpf_coarse_kernel(
    const float* __restrict__ pts, const float* __restrict__ feat,
    const float* __restrict__ lv,  const int* __restrict__ mask,
    const float* __restrict__ W1,  const float* __restrict__ bi1,
    const float* __restrict__ W2,  const float* __restrict__ bi2,
    const float* __restrict__ w3v, const float* __restrict__ b3v,
    float* __restrict__ scores, float* __restrict__ outc)
{
    __shared__ float xs[NT][XS_STRIDE];   // point-major padded input tile
    __shared__ float hs[NT][HS_STRIDE];   // point-major layer-1 activations
    __shared__ float ss[NT];              // per-point score accumulator

    const int b     = blockIdx.y;
    const int pbase = blockIdx.x * NT;
    const int tid   = threadIdx.x;
    const int w     = tid >> 5;           // wave id 0..7 -> M-tile
    const int L     = tid & 31;
    const int l15   = L & 15;
    const int hi    = L >> 4;             // lane half
    const int koff  = hi * 2;             // K offset for A/B operands
    const int mhalf = hi * 8;             // M offset for C/D rows

    // zero the two padded K columns used by the k-loop
    for (int i = tid; i < NT * 2; i += 256)
        xs[i >> 1][CIN + (i & 1)] = 0.f;

    // async DMA transpose staging: per-lane global->LDS scatter, tracked by ASYNCcnt
    for (int i = tid; i < CIN * NT; i += 256) {
        int c = i / NT, n = i % NT;       // consecutive lanes -> consecutive global addrs
        int p = pbase + n;
        const float* src;
        if (c < 2)       src = &pts [((size_t)b * 2  + c)        * P_SZ + p];
        else if (c < 18) src = &feat[((size_t)b * 16 + (c - 2))  * P_SZ + p];
        else             src = &lv  [((size_t)b * 4  + (c - 18)) * P_SZ + p];
        unsigned ldsoff = (unsigned)(unsigned long long)&xs[n][c];
        asm volatile("global_load_async_to_lds_b32 %0, %1, off"
                     :: "v"(ldsoff), "v"((unsigned long long)src) : "memory");
    }
    if (tid < NT) ss[tid] = b3v[0];
    ttpf_wait_async0();                   // drain ASYNCcnt before the barrier
    __syncthreads();

    const int m_base = w * 16;
    const int m      = m_base + l15;

    __builtin_prefetch(&W2[m * HC], 0, 1);

    // ---- layer 1: h = relu(W1 @ x + b1) ----
    v8f acc[NTILES] = {};
    for (int ks = 0; ks < CPAD / 4; ++ks) {
        int k0 = ks * 4 + koff;
        int kc = (k0 + 1 < CIN) ? k0 : 0;             // branch-free edge handling
        v2f av = *(const v2f*)&W1[m * CIN + kc];
        v2f a;
        a.x = (k0     < CIN) ? av.x : 0.f;
        a.y = (k0 + 1 < CIN) ? av.y : 0.f;
        for (int t = 0; t < NTILES; ++t) {
            v2f bb = *(const v2f*)&xs[t * 16 + l15][k0];     // one ds_load_b64
            acc[t] = __builtin_amdgcn_wmma_f32_16x16x4_f32(
                false, a, false, bb, (short)0, acc[t], false, false);
        }
    }
    for (int t = 0; t < NTILES; ++t) {
        v4f h0, h1;
        for (int j = 0; j < 4; ++j) {
            h0[j] = fmaxf(acc[t][j]     + bi1[m_base + mhalf + j],     0.f);
            h1[j] = fmaxf(acc[t][j + 4] + bi1[m_base + mhalf + j + 4], 0.f);
        }
        *(v4f*)&hs[t * 16 + l15][m_base + mhalf]     = h0;   // ds_store_b128
        *(v4f*)&hs[t * 16 + l15][m_base + mhalf + 4] = h1;
    }
    __syncthreads();

    // ---- layer 2 + score: s = w3 . relu(W2 @ h + b2) + b3 ----
    v8f acc2[NTILES] = {};
    for (int ks = 0; ks < HC / 4; ++ks) {
        int k0 = ks * 4 + koff;
        v2f a = *(const v2f*)&W2[m * HC + k0];                // global_load_b64
        for (int t = 0; t < NTILES; ++t) {
            v2f bb = *(const v2f*)&hs[t * 16 + l15][k0];      // ds_load_b64
            acc2[t] = __builtin_amdgcn_wmma_f32_16x16x4_f32(
                false, a, false, bb, (short)0, acc2[t], false, false);
        }
    }
    for (int t = 0; t < NTILES; ++t) {
        float part = 0.f;
        for (int j = 0; j < 8; ++j) {
            int mm = m_base + j + mhalf;
            part += w3v[mm] * fmaxf(acc2[t][j] + bi2[mm], 0.f);
        }
        atomicAdd(&ss[t * 16 + l15], part);                   // ds_add_f32
    }
    __syncthreads();

    if (tid < NT) {
        int p = pbase + tid;
        bool valid = mask[(size_t)b * P_SZ + p] > 0;
        float s = ss[tid];
        scores[(size_t)b * P_SZ + p] = valid ? s : NEGINF;
        outc  [(size_t)b * P_SZ + p] = valid ? (s - OFF_V) : NEGINF;
    }
}

// ---------------- exact top-600 per batch via 4-pass radix select ----------------
__device__ __forceinline__ unsigned ttpf_key(float f) {
    unsigned u = __float_as_uint(f);
    return u ^ ((u & 0x80000000u) ? 0xFFFFFFFFu : 0x80000000u);
}

__global__ __launch_bounds__(256) void ttpf_topk_kernel(
    const float* __restrict__ scores, int* __restrict__ idx_out)
{
    __shared__ unsigned hist[256];
    __shared__ unsigned s_prefix, s_remaining;
    __shared__ unsigned cntAbove, cntEq;

    const int b   = blockIdx.x;
    const int tid = threadIdx.x;
    const float* sc = scores + (size_t)b * P_SZ;

    if (tid == 0) { s_prefix = 0u; s_remaining = TOPN; }
    unsigned maskSoFar = 0u;
    __syncthreads();

    for (int pass = 0; pass < 4; ++pass) {
        int shift = 24 - 8 * pass;
        hist[tid] = 0u;
        __syncthreads();
        unsigned prefix = s_prefix;
        for (int p = tid; p < P_SZ; p += 256) {
            unsigned key = ttpf_key(sc[p]);
            if ((key & maskSoFar) == prefix)
                atomicAdd(&hist[(key >> shift) & 255u], 1u);
        }
        __syncthreads();
        if (tid == 0) {
            unsigned rem = s_remaining, accum = 0u;
            int binSel = 0;
            for (int bin = 255; bin >= 0; --bin) {
                accum += hist[bin];
                if (accum >= rem) { binSel = bin; rem -= (accum - hist[bin]); break; }
            }
            s_prefix    = prefix | ((unsigned)binSel << shift);
            s_remaining = rem;
        }
        maskSoFar |= (0xFFu << shift);
        __syncthreads();
    }

    const unsigned T     = s_prefix;     // full 32-bit threshold key
    const unsigned numEq = s_remaining;  // #ties at T still needed
    if (tid == 0) { cntAbove = 0u; cntEq = 0u; }
    __syncthreads();

    int* dst = idx_out + (size_t)b * TOPN;
    for (int p = tid; p < P_SZ; p += 256) {
        if (ttpf_key(sc[p]) > T) {
            unsigned pos = atomicAdd(&cntAbove, 1u);
            dst[pos] = p;
        }
    }
    __syncthreads();
    const unsigned nAbove = cntAbove;    // == TOPN - numEq
    for (int p = tid; p < P_SZ; p += 256) {
        if (ttpf_key(sc[p]) == T) {
            unsigned e = atomicAdd(&cntEq, 1u);
            if (e < numEq) dst[nAbove + e] = p;
        }
    }
}

// ---------------- refine MLP: 22 -> 384 -> 384 -> 1 over selected points ----------------
#define NTR        32                     // selected points per block
#define MT_R       3                      // M-tiles per wave (24 tiles / 8 waves)
#define NBLK_R     ((TOPN + NTR - 1) / NTR)
#define HSR_STRIDE (HR + 4)               // 388 : 16B-aligned, conflict-free

__global__ __launch_bounds__(256) void ttpf_refine_kernel(
    const float* __restrict__ pts, const float* __restrict__ feat,
    const float* __restrict__ lv,  const int* __restrict__ mask,
    const float* __restrict__ W1,  const float* __restrict__ bi1,
    const float* __restrict__ W2,  const float* __restrict__ bi2,
    const float* __restrict__ w3v, const float* __restrict__ b3v,
    const int* __restrict__ idx,   float* __restrict__ outc)
{
    __shared__ float xs[NTR][XS_STRIDE];  // 3.5 KB, point-major
    __shared__ float hs[NTR][HSR_STRIDE]; // 48.5 KB, point-major
    __shared__ float ss[NTR];
    __shared__ int   sidx[NTR];

    const int b     = blockIdx.y;
    const int sbase = blockIdx.x * NTR;
    const int tid   = threadIdx.x;
    const int w     = tid >> 5;
    const int L     = tid & 31;
    const int l15   = L & 15;
    const int hi    = L >> 4;
    const int koff  = hi * 2;
    const int mhalf = hi * 8;

    if (tid < NTR) {
        int s = sbase + tid;
        sidx[tid] = (s < TOPN) ? idx[(size_t)b * TOPN + s] : -1;
        ss[tid] = b3v[0];
    }
    __syncthreads();

    // gather selected x columns (point-major, cols 22..23 zeroed)
    for (int i = tid; i < NTR * CPAD; i += 256) {
        int n = i / CPAD, c = i % CPAD;
        int p = sidx[n];
        float v = 0.f;
        if (p >= 0 && c < CIN) {
            if (c < 2)       v = pts [((size_t)b * 2  + c)        * P_SZ + p];
            else if (c < 18) v = feat[((size_t)b * 16 + (c - 2))  * P_SZ + p];
            else             v = lv  [((size_t)b * 4  + (c - 18)) * P_SZ + p];
        }
        xs[n][c] = v;
    }
    __syncthreads();

    __builtin_prefetch(&W2[(w * 16 + l15) * HR], 0, 1);

    // ---- layer 1 ----
    for (int mt = 0; mt < MT_R; ++mt) {
        const int m_base = (w + 8 * mt) * 16;
        const int m      = m_base + l15;
        v8f acc0 = {}, acc1 = {};
        for (int ks = 0; ks < CPAD / 4; ++ks) {
            int k0 = ks * 4 + koff;
            int kc = (k0 + 1 < CIN) ? k0 : 0;         // branch-free edge handling
            v2f av = *(const v2f*)&W1[m * CIN + kc];
            v2f a;
            a.x = (k0     < CIN) ? av.x : 0.f;
            a.y = (k0 + 1 < CIN) ? av.y : 0.f;
            v2f bb0 = *(const v2f*)&xs[l15     ][k0];
            v2f bb1 = *(const v2f*)&xs[16 + l15][k0];
            acc0 = __builtin_amdgcn_wmma_f32_16x16x4_f32(false, a, false, bb0, (short)0, acc0, false, false);
            acc1 = __builtin_amdgcn_wmma_f32_16x16x4_f32(false, a, false, bb1, (short)0, acc1, false, false);
        }
        v4f h00, h01, h10, h11;
        for (int j = 0; j < 4; ++j) {
            int mm = m_base + mhalf + j;
            h00[j] = fmaxf(acc0[j]     + bi1[mm],     0.f);
            h01[j] = fmaxf(acc0[j + 4] + bi1[mm + 4], 0.f);
            h10[j] = fmaxf(acc1[j]     + bi1[mm],     0.f);
            h11[j] = fmaxf(acc1[j + 4] + bi1[mm + 4], 0.f);
        }
        *(v4f*)&hs[l15     ][m_base + mhalf]     = h00;
        *(v4f*)&hs[l15     ][m_base + mhalf + 4] = h01;
        *(v4f*)&hs[16 + l15][m_base + mhalf]     = h10;
        *(v4f*)&hs[16 + l15][m_base + mhalf + 4] = h11;
    }
    __syncthreads();

    // ---- layer 2 + score ----
    for (int mt = 0; mt < MT_R; ++mt) {
        const int m_base = (w + 8 * mt) * 16;
        const int m      = m_base + l15;
        v8f acc0 = {}, acc1 = {};
        for (int ks = 0; ks < HR / 4; ++ks) {
            int k0 = ks * 4 + koff;
            v2f a   = *(const v2f*)&W2[m * HR + k0];
            v2f bb0 = *(const v2f*)&hs[l15     ][k0];
            v2f bb1 = *(const v2f*)&hs[16 + l15][k0];
            acc0 = __builtin_amdgcn_wmma_f32_16x16x4_f32(false, a, false, bb0, (short)0, acc0, false, false);
            acc1 = __builtin_amdgcn_wmma_f32_16x16x4_f32(false, a, false, bb1, (short)0, acc1, false, false);
        }
        float p0 = 0.f, p1 = 0.f;
        for (int j = 0; j < 8; ++j) {
            int mm = m_base + j + mhalf;
            p0 += w3v[mm] * fmaxf(acc0[j] + bi2[mm], 0.f);
            p1 += w3v[mm] * fmaxf(acc1[j] + bi2[mm], 0.f);
        }
        atomicAdd(&ss[l15],      p0);
        atomicAdd(&ss[16 + l15], p1);
    }
    __syncthreads();

    if (tid < NTR) {
        int p = sidx[tid];
        if (p >= 0 && mask[(size_t)b * P_SZ + p] > 0)
            outc[(size_t)b * P_SZ + p] = ss[tid];
    }
}

// ---------------- launcher ----------------
extern "C" void kernel_launch(void* const* d_in, const int* in_sizes, int n_in,
                              void* d_out, int out_size, void* d_ws, size_t ws_size,
                              hipStream_t stream)
{
    const float* pts  = (const float*)d_in[0];
    const float* feat = (const float*)d_in[1];
    const float* lv   = (const float*)d_in[2];
    const int*   mask = (const int*)  d_in[3];
    const float* Wc1  = (const float*)d_in[4];
    const float* bc1  = (const float*)d_in[5];
    const float* Wc2  = (const float*)d_in[6];
    const float* bc2  = (const float*)d_in[7];
    const float* wc3  = (const float*)d_in[8];
    const float* bc3  = (const float*)d_in[9];
    const float* Wr1  = (const float*)d_in[10];
    const float* br1  = (const float*)d_in[11];
    const float* Wr2  = (const float*)d_in[12];
    const float* br2  = (const float*)d_in[13];
    const float* wr3  = (const float*)d_in[14];
    const float* br3  = (const float*)d_in[15];

    float* out       = (float*)d_out;
    float* ws_scores = (float*)d_ws;                                              // B*P floats
    int*   ws_idx    = (int*)((char*)d_ws + (size_t)B_SZ * P_SZ * sizeof(float)); // B*TOPN ints

    ttpf_coarse_kernel<<<dim3(P_SZ / NT, B_SZ), 256, 0, stream>>>(
        pts, feat, lv, mask, Wc1, bc1, Wc2, bc2, wc3, bc3, ws_scores, out);

    ttpf_topk_kernel<<<B_SZ, 256, 0, stream>>>(ws_scores, ws_idx);

    ttpf_refine_kernel<<<dim3(NBLK_R, B_SZ), 256, 0, stream>>>(
        pts, feat, lv, mask, Wr1, br1, Wr2, br2, wr3, br3, ws_idx, out);
}